// LocalGOCor_20572893348666
// MI455X (gfx1250) — compile-verified
//
#include <hip/hip_runtime.h>

// LocalGOCor optimizer for MI455X (gfx1250), fp32 end-to-end.
// All cost volumes and 81->32 projections run on V_WMMA_F32_16X16X4_F32.
// Round 2: branch-free masked loads (no saveexec serialization), batched
// fragment loads with dual interleaved WMMA accumulator chains, projection
// weights staged in LDS (zero-padded), 16B-aligned LDS pitches.
// Workspace requirement: 3 * B*H*W*C floats + 1KB  (~101 MB).

#define B_  4
#define H_  256
#define W_  256
#define C_  32
#define K81 81
#define CBW 88    // cbuf row pitch (floats): 81 + pad; 88*4 B = 16B-aligned rows
#define STG 33    // stage row pitch (16 rows x 32 cols + conflict pad)
#define NW  4     // waves per block (128 threads)

typedef float v2f __attribute__((ext_vector_type(2)));
typedef float v8f __attribute__((ext_vector_type(8)));

static __device__ __forceinline__ v8f wmma4(v2f a, v2f b, v8f c) {
  // D(16x16,f32) += A(16x4,f32) x B(4x16,f32)
  return __builtin_amdgcn_wmma_f32_16x16x4_f32(false, a, false, b, (short)0, c,
                                               false, false);
}

static __device__ __forceinline__ int iclamp(int v, int lo, int hi) {
  return v < lo ? lo : (v > hi ? hi : v);
}

// One wave computes cbuf[16][CBW] = local cost volume (mean over C) for
// pixels (b, y, x0..x0+15), window 9x9, via WMMA GEMMs staged through LDS.
static __device__ void wave_costvol(const float* __restrict__ f1,
                                    const float* __restrict__ ref,
                                    int b, int y, int x0, int lane,
                                    float* stage, float* cbuf) {
  const int m = lane & 15, half = lane >> 4;
  const int chb = 2 * half;
  // A fragments: rows = pixels, K = channels (preload all 8 K-slabs)
  v2f afrag[8];
  const float* arow = f1 + (size_t)((b * H_ + y) * W_ + x0 + m) * C_;
#pragma unroll
  for (int kb = 0; kb < 8; ++kb) {
    afrag[kb].x = arow[kb * 4 + chb];
    afrag[kb].y = arow[kb * 4 + chb + 1];
  }
  for (int di = 0; di < 9; ++di) {
    int yy = y + di - 4;
    int yyc = iclamp(yy, 0, H_ - 1);
    float rowmk = (yy >= 0 && yy < H_) ? 1.f : 0.f;
    // Two 16-wide q-tiles cover q = x0-4 .. x0+27 (band for all 9 dj shifts).
    int q0 = x0 - 4 + m;
    int q1 = q0 + 16;                              // always >= 12
    float mk0 = rowmk * ((q0 >= 0 && q0 < W_) ? 1.f : 0.f);
    float mk1 = rowmk * ((q1 < W_) ? 1.f : 0.f);
    const float* brow0 =
        ref + (size_t)((b * H_ + yyc) * W_ + iclamp(q0, 0, W_ - 1)) * C_;
    const float* brow1 =
        ref + (size_t)((b * H_ + yyc) * W_ + (q1 < W_ ? q1 : W_ - 1)) * C_;
    // Batch all 16 b64 loads (branch-free), then mask-scale.
    v2f bf0[8], bf1[8];
#pragma unroll
    for (int kb = 0; kb < 8; ++kb) {
      int ch = kb * 4 + chb;
      bf0[kb].x = brow0[ch] * mk0;  bf0[kb].y = brow0[ch + 1] * mk0;
      bf1[kb].x = brow1[ch] * mk1;  bf1[kb].y = brow1[ch + 1] * mk1;
    }
    // Two independent accumulator chains, interleaved to hide WMMA latency.
    v8f acc0 = {}, acc1 = {};
#pragma unroll
    for (int kb = 0; kb < 8; ++kb) {
      acc0 = wmma4(afrag[kb], bf0[kb], acc0);
      acc1 = wmma4(afrag[kb], bf1[kb], acc1);
    }
    // D[row][col]: vgpr r -> row = r + 8*half, col = tile*16 + m
#pragma unroll
    for (int r = 0; r < 8; ++r) {
      stage[(r + 8 * half) * STG + m]      = acc0[r];
      stage[(r + 8 * half) * STG + 16 + m] = acc1[r];
    }
    __builtin_amdgcn_wave_barrier();
    if (half == 0) {  // pixel m gathers its diagonal band: col = m + dj
#pragma unroll
      for (int dj = 0; dj < 9; ++dj)
        cbuf[m * CBW + di * 9 + dj] = stage[m * STG + m + dj] * (1.0f / 32.0f);
    }
    __builtin_amdgcn_wave_barrier();
  }
}

// acc0/acc1 = cbuf[16][84] @ wlds[84][32] (two 16-col output tiles).
// wlds is the 81x32 weight matrix staged in LDS, zero-padded to 84 rows.
static __device__ void wave_proj(const float* cbuf, const float* wlds,
                                 int lane, v8f* o0, v8f* o1) {
  const int m = lane & 15, half = lane >> 4;
  v8f a0 = {}, a1 = {};
#pragma unroll
  for (int kb = 0; kb < 84; kb += 4) {
    int k0 = kb + 2 * half;
    v2f a;  a.x  = cbuf[m * CBW + k0];        a.y  = cbuf[m * CBW + k0 + 1];
    v2f b0; b0.x = wlds[k0 * C_ + m];         b0.y = wlds[(k0 + 1) * C_ + m];
    v2f b1; b1.x = wlds[k0 * C_ + 16 + m];    b1.y = wlds[(k0 + 1) * C_ + 16 + m];
    a0 = wmma4(a, b0, a0);
    a1 = wmma4(a, b1, a1);
  }
  *o0 = a0; *o1 = a1;
}

// ---------------- small precompute kernels ----------------

// cst layout: y[0..80], v_plus[81..161], weight_m[162..242], reg_w[243], step[244]
__global__ void lg_consts(const float* __restrict__ tmw, const float* __restrict__ spw,
                          const float* __restrict__ mkw, const float* __restrict__ wreg,
                          const float* __restrict__ lsl, float* __restrict__ cst) {
  int t = threadIdx.x;
  if (t < K81) {
    int i = t / 9, j = t % 9;
    float di = (float)(i - 4), dj = (float)(j - 4);
    float dist = sqrtf(di * di + dj * dj);
    float yv = 0.f, vv = 0.f, mv = 0.f;
#pragma unroll
    for (int bn = 0; bn < 10; ++bn) {
      float bd = dist * 2.0f - (float)bn;  // dist / BIN_DISP - bin
      float bv = (i < 8) ? fmaxf(1.f - fabsf(bd), 0.f)
                         : fminf(fmaxf(1.f + bd, 0.f), 1.f);
      yv += bv * tmw[bn]; vv += bv * spw[bn]; mv += bv * mkw[bn];
    }
    cst[t] = yv;
    cst[K81 + t] = vv;
    cst[2 * K81 + t] = 1.f / (1.f + expf(-mv));
  }
  if (t == K81) {
    float r = wreg[0] * wreg[0];
    cst[243] = fmaxf(r, 9.765625e-14f);  // (1e-5)^2 / 32^2
    cst[244] = expf(lsl[0]);
  }
}

__global__ void lg_init_w(const float* __restrict__ ref,
                          const float* __restrict__ beta,
                          float* __restrict__ w) {
  size_t p = (size_t)blockIdx.x * blockDim.x + threadIdx.x;
  if (p >= (size_t)B_ * H_ * W_) return;
  const float* r = ref + p * C_;
  float ss = 0.f;
#pragma unroll
  for (int c = 0; c < C_; ++c) ss += r[c] * r[c];
  float scale = beta[0] / (ss * (1.0f / C_) + 1e-6f);
  float* o = w + p * C_;
#pragma unroll
  for (int c = 0; c < C_; ++c) o[c] = r[c] * scale;
}

// ---------------- iteration kernels ----------------

// A: c = costvol(w, ref); E = dact*res; mapped = E @ mr_w + mr_b
__global__ __launch_bounds__(128) void lg_kernelA(
    const float* __restrict__ wcur, const float* __restrict__ ref,
    const float* __restrict__ mrw, const float* __restrict__ mrb,
    const float* __restrict__ cst, float* __restrict__ mapped) {
  __shared__ __align__(16) float stage[NW][16 * STG];
  __shared__ __align__(16) float cbuf[NW][16 * CBW];
  __shared__ __align__(16) float wlds[84 * C_];
  __shared__ float s_y[K81], s_vp[K81], s_wm[K81];
  int tid = threadIdx.x;
  for (int i = tid; i < K81; i += blockDim.x) {
    s_y[i] = cst[i]; s_vp[i] = cst[K81 + i]; s_wm[i] = cst[2 * K81 + i];
  }
  for (int i = tid; i < 84 * C_; i += blockDim.x)
    wlds[i] = (i < K81 * C_) ? mrw[i] : 0.f;
  __syncthreads();
  int wave = tid >> 5, lane = tid & 31;
  int b = blockIdx.z, y = blockIdx.y, x0 = blockIdx.x * 64 + wave * 16;

  wave_costvol(wcur, ref, b, y, x0, lane, stage[wave], cbuf[wave]);

  // c -> dact*res in place
  for (int idx = lane; idx < 16 * K81; idx += 32) {
    int mm = idx / K81, k = idx % K81;
    float c = cbuf[wave][mm * CBW + k];
    float a = s_wm[k], vp = s_vp[k];
    float lp = 0.5f * (1.f - a) * fabsf(c) + 0.5f * (1.f + a) * c;
    float sg = (c > 0.f) ? 1.f : ((c < 0.f) ? -1.f : 0.f);
    float act = vp * lp;
    float dact = vp * (0.5f * (1.f - a) * sg + 0.5f * (1.f + a));
    float res = act - vp * s_y[k];
    cbuf[wave][mm * CBW + k] = dact * res;
  }
  int m = lane & 15, half = lane >> 4;
  if (half == 0) {  // zero-pad K to 84
    cbuf[wave][m * CBW + 81] = 0.f;
    cbuf[wave][m * CBW + 82] = 0.f;
    cbuf[wave][m * CBW + 83] = 0.f;
  }
  __builtin_amdgcn_wave_barrier();

  v8f a0, a1;
  wave_proj(cbuf[wave], wlds, lane, &a0, &a1);
#pragma unroll
  for (int r = 0; r < 8; ++r) {
    int row = r + 8 * half;
    size_t base = (size_t)((b * H_ + y) * W_ + x0 + row) * C_;
    mapped[base + m]      = a0[r] + mrb[m];
    mapped[base + 16 + m] = a1[r] + mrb[16 + m];
  }
}

// B: fg = costvol(mapped, ref) @ lr_w + lr_b + reg_w * w
__global__ __launch_bounds__(128) void lg_kernelB(
    const float* __restrict__ wcur, const float* __restrict__ mapped,
    const float* __restrict__ ref, const float* __restrict__ lrw,
    const float* __restrict__ lrb, const float* __restrict__ cst,
    float* __restrict__ fg) {
  __shared__ __align__(16) float stage[NW][16 * STG];
  __shared__ __align__(16) float cbuf[NW][16 * CBW];
  __shared__ __align__(16) float wlds[84 * C_];
  int tid = threadIdx.x;
  for (int i = tid; i < 84 * C_; i += blockDim.x)
    wlds[i] = (i < K81 * C_) ? lrw[i] : 0.f;
  __syncthreads();
  int wave = tid >> 5, lane = tid & 31;
  int b = blockIdx.z, y = blockIdx.y, x0 = blockIdx.x * 64 + wave * 16;
  float reg_w = cst[243];

  wave_costvol(mapped, ref, b, y, x0, lane, stage[wave], cbuf[wave]);

  int m = lane & 15, half = lane >> 4;
  if (half == 0) {
    cbuf[wave][m * CBW + 81] = 0.f;
    cbuf[wave][m * CBW + 82] = 0.f;
    cbuf[wave][m * CBW + 83] = 0.f;
  }
  __builtin_amdgcn_wave_barrier();

  v8f a0, a1;
  wave_proj(cbuf[wave], wlds, lane, &a0, &a1);
#pragma unroll
  for (int r = 0; r < 8; ++r) {
    int row = r + 8 * half;
    size_t base = (size_t)((b * H_ + y) * W_ + x0 + row) * C_;
    fg[base + m]      = a0[r] + lrb[m]      + reg_w * wcur[base + m];
    fg[base + 16 + m] = a1[r] + lrb[16 + m] + reg_w * wcur[base + 16 + m];
  }
}

// C: s0 = costvol(fg, ref); dact from recomputed costvol(w, ref);
//    alpha = sum(fg^2) / sum((dact*s0)^2);  w_next = w + alpha*step*fg
__global__ __launch_bounds__(128) void lg_kernelC(
    const float* __restrict__ wcur, const float* __restrict__ fg,
    const float* __restrict__ ref, const float* __restrict__ cst,
    float* __restrict__ wnext) {
  __shared__ __align__(16) float stage[NW][16 * STG];
  __shared__ __align__(16) float cbufA[NW][16 * CBW];  // c -> dact
  __shared__ __align__(16) float cbufB[NW][16 * CBW];  // s0
  __shared__ float s_vp[K81], s_wm[K81];
  int tid = threadIdx.x;
  for (int i = tid; i < K81; i += blockDim.x) {
    s_vp[i] = cst[K81 + i]; s_wm[i] = cst[2 * K81 + i];
  }
  __syncthreads();
  int wave = tid >> 5, lane = tid & 31;
  int b = blockIdx.z, y = blockIdx.y, x0 = blockIdx.x * 64 + wave * 16;
  float step = cst[244];

  wave_costvol(wcur, ref, b, y, x0, lane, stage[wave], cbufA[wave]);
  wave_costvol(fg,   ref, b, y, x0, lane, stage[wave], cbufB[wave]);

  int m = lane & 15, half = lane >> 4;
  if (half == 0) {
    float den = 0.f;
    for (int k = 0; k < K81; ++k) {
      float c = cbufA[wave][m * CBW + k];
      float a = s_wm[k], vp = s_vp[k];
      float sg = (c > 0.f) ? 1.f : ((c < 0.f) ? -1.f : 0.f);
      float dact = vp * (0.5f * (1.f - a) * sg + 0.5f * (1.f + a));
      float sc = dact * cbufB[wave][m * CBW + k];
      den += sc * sc;
    }
    size_t base = (size_t)((b * H_ + y) * W_ + x0 + m) * C_;
    float num = 0.f;
#pragma unroll
    for (int c = 0; c < C_; ++c) { float g = fg[base + c]; num += g * g; }
    float as = (num / den) * step;
#pragma unroll
    for (int c = 0; c < C_; ++c) wnext[base + c] = wcur[base + c] + as * fg[base + c];
  }
}

// ---------------- launcher ----------------

extern "C" void kernel_launch(void* const* d_in, const int* in_sizes, int n_in,
                              void* d_out, int out_size, void* d_ws, size_t ws_size,
                              hipStream_t stream) {
  const float* ref  = (const float*)d_in[0];   // reference_feature (B,H,W,C)
  // d_in[1] = query_feature (unused by the reference computation)
  const float* beta = (const float*)d_in[2];
  const float* tmw  = (const float*)d_in[3];
  const float* spw  = (const float*)d_in[4];
  const float* mkw  = (const float*)d_in[5];
  const float* mrw  = (const float*)d_in[6];
  const float* mrb  = (const float*)d_in[7];
  const float* lrw  = (const float*)d_in[8];
  const float* lrb  = (const float*)d_in[9];
  const float* wrg  = (const float*)d_in[10];
  const float* lsl  = (const float*)d_in[11];
  float* out = (float*)d_out;

  const size_t N = (size_t)B_ * H_ * W_ * C_;
  float* wsf    = (float*)d_ws;
  float* cst    = wsf;            // 245 floats (reserve 256)
  float* w_a    = wsf + 256;      // w double-buffer half
  float* mapped = w_a + N;
  float* fgbuf  = mapped + N;

  lg_consts<<<1, 128, 0, stream>>>(tmw, spw, mkw, wrg, lsl, cst);
  lg_init_w<<<(B_ * H_ * W_ + 255) / 256, 256, 0, stream>>>(ref, beta, w_a);

  dim3 grid(W_ / 64, H_, B_);
  dim3 blk(128);
  float* cur = w_a;   // iter parity: w_a -> out -> w_a -> out (3 iters)
  float* nxt = out;
  for (int it = 0; it < 3; ++it) {
    lg_kernelA<<<grid, blk, 0, stream>>>(cur, ref, mrw, mrb, cst, mapped);
    lg_kernelB<<<grid, blk, 0, stream>>>(cur, mapped, ref, lrw, lrb, cst, fgbuf);
    lg_kernelC<<<grid, blk, 0, stream>>>(cur, fgbuf, ref, cst, nxt);
    float* t = cur; cur = nxt; nxt = t;
  }
}